// NTXentLossChildrenFmaps2_85624468013609
// MI455X (gfx1250) — compile-verified
//
#include <hip/hip_runtime.h>
#include <math.h>

// ---------------- problem constants (fixed by the reference) ----------------
#define BDIM 8
#define JDIM 4
#define KDIM 3
#define DDIM 32768
#define NDIM 4
#define MDIM 16
#define NROWS (NDIM * MDIM * KDIM)   // 192 neg rows per batch
#define NTILES (NROWS / 16)          // 12 tiles of 16 rows
#define TEMP_INV 2.0f
#define EPSV 1e-8f

// ---------------- workspace layout (floats) ----------------
#define XDOT_OFF 0                        // 24: dot(x,xp)
#define XN2_OFF 24                        // 24: ||x||^2
#define XPN2_OFF 48                       // 24: ||xp||^2
#define DOT_OFF 72                        // [b][tile][r(0..2)][n(0..15)] = 4608
#define NEG2_OFF (72 + 4608)              // [b][g(0..191)] = 1536
#define WS_FLOATS (NEG2_OFF + BDIM * NROWS)

// LDS row stride for the staged A rows: 2048 + 4 pad -> rows start on
// banks 0/4/8, so lanes reading rows 0..2 at the same d-offset never conflict.
#define A_STRIDE 2052

typedef float v2f __attribute__((ext_vector_type(2)));
typedef float v8f __attribute__((ext_vector_type(8)));

// ---------------- 0: zero the accumulators ----------------
__global__ __launch_bounds__(256) void zero_ws_kernel(float* __restrict__ ws) {
    for (int i = threadIdx.x; i < WS_FLOATS; i += 256) ws[i] = 0.0f;
}

// ---------------- 1: pos dots + x / xp norms ----------------
__global__ __launch_bounds__(256) void pos_stats_kernel(const float* __restrict__ fmaps,
                                                        const float* __restrict__ fmaps_pos,
                                                        float* __restrict__ ws) {
    const int pair = blockIdx.x;              // 0..23 -> (b, k)
    const int b = pair / KDIM;
    const int k = pair % KDIM;
    const size_t row = ((size_t)(b * JDIM + (JDIM - 1)) * KDIM + k) * DDIM;
    const float4* __restrict__ x4  = (const float4*)(fmaps + row);
    const float4* __restrict__ xp4 = (const float4*)(fmaps_pos + row);

    float dp = 0.f, nx = 0.f, np = 0.f;
    for (int i = threadIdx.x; i < DDIM / 4; i += 256) {
        float4 a = x4[i];
        float4 p = xp4[i];
        dp += a.x * p.x + a.y * p.y + a.z * p.z + a.w * p.w;
        nx += a.x * a.x + a.y * a.y + a.z * a.z + a.w * a.w;
        np += p.x * p.x + p.y * p.y + p.z * p.z + p.w * p.w;
    }
    __shared__ float s0[256], s1[256], s2[256];
    const int tid = threadIdx.x;
    s0[tid] = dp; s1[tid] = nx; s2[tid] = np;
    __syncthreads();
    for (int off = 128; off > 0; off >>= 1) {
        if (tid < off) {
            s0[tid] += s0[tid + off];
            s1[tid] += s1[tid + off];
            s2[tid] += s2[tid + off];
        }
        __syncthreads();
    }
    if (tid == 0) {
        ws[XDOT_OFF + pair] = s0[0];
        ws[XN2_OFF  + pair] = s1[0];
        ws[XPN2_OFF + pair] = s2[0];
    }
}

// ---------------- 2: WMMA GEMM over fmaps_neg + row norms ----------------
// grid: (b=8, tile=12, dchunk=16) ; block: 128 threads = 4 waves.
// A (3 x-rows, 2048 floats of d per block) is staged into LDS once via
// async global->LDS copies (ASYNCcnt). The hot loop batches 8 K-steps:
// 8 independent global_load_b64 of B (2 KB in flight per wave) + 8
// ds_load_b64 of A, then 8 chained v_wmma_f32_16x16x4_f32.
__global__ __launch_bounds__(128) void neg_gemm_kernel(const float* __restrict__ fmaps,
                                                       const float* __restrict__ fmaps_neg,
                                                       float* __restrict__ ws) {
    const int b    = blockIdx.x;
    const int tile = blockIdx.y;
    const int tid  = threadIdx.x;
    const int wave = tid >> 5;
    const int lane = tid & 31;
    const int half = lane >> 4;       // selects K pair {2*half, 2*half+1}
    const int n16  = lane & 15;       // A: row M ; B: column N (neg row in tile)
    const int dblk   = blockIdx.z * 2048;     // block's d-range
    const int dstart = dblk + wave * 512;     // this wave's d-range

    __shared__ float lds_a[3 * A_STRIDE];

    const float* __restrict__ xbase =
        fmaps + (size_t)(b * JDIM + (JDIM - 1)) * KDIM * DDIM;

    // ---- stage A rows into LDS with async copies (uniform, no divergence)
    for (int i = tid; i < 3 * 512; i += 128) {          // 1536 float4 tiles
        const int r  = i / 512;                          // x row 0..2
        const int c4 = i % 512;                          // float4 column
        const float* gsrc = xbase + (size_t)r * DDIM + dblk + c4 * 4;
        unsigned ldst = (unsigned)(uintptr_t)(&lds_a[r * A_STRIDE + c4 * 4]);
        unsigned long long gaddr = (unsigned long long)(uintptr_t)gsrc;
        asm volatile("global_load_async_to_lds_b128 %0, %1, off"
                     :: "v"(ldst), "v"(gaddr) : "memory");
    }
    asm volatile("s_wait_asynccnt 0x0" ::: "memory");
    __syncthreads();

    const int arowidx = (n16 < KDIM) ? n16 : 0;
    const float amask = (n16 < KDIM) ? 1.0f : 0.0f;
    const float* __restrict__ lds_arow = &lds_a[arowidx * A_STRIDE];

    const int g = tile * 16 + n16;    // global neg row 0..191
    const float* __restrict__ negrow = fmaps_neg + ((size_t)b * NROWS + g) * DDIM;

    v8f c = {0.f, 0.f, 0.f, 0.f, 0.f, 0.f, 0.f, 0.f};
    float nrm = 0.f;

    for (int dseg = dstart; dseg < dstart + 512; dseg += 32) {
        v2f bv8[8], av8[8];
#pragma unroll
        for (int u = 0; u < 8; ++u)       // 8 independent B loads -> deep MLP
            bv8[u] = *(const v2f*)(negrow + dseg + 4 * u + 2 * half);
#pragma unroll
        for (int u = 0; u < 8; ++u)       // A from LDS (ds_load_b64)
            av8[u] = *(const v2f*)(lds_arow + (dseg - dblk) + 4 * u + 2 * half);
        if (dseg + 4096 < DDIM)
            __builtin_prefetch(negrow + dseg + 4096, 0, 3);   // global_prefetch_b8
#pragma unroll
        for (int u = 0; u < 8; ++u) {
            v2f a;
            a.x = av8[u].x * amask;
            a.y = av8[u].y * amask;
            nrm = __builtin_fmaf(bv8[u].x, bv8[u].x, nrm);
            nrm = __builtin_fmaf(bv8[u].y, bv8[u].y, nrm);
            // D = A(16x4) * B(4x16) + C  -> v_wmma_f32_16x16x4_f32
            c = __builtin_amdgcn_wmma_f32_16x16x4_f32(
                    /*neg_a=*/false, a, /*neg_b=*/false, bv8[u],
                    /*c_mod=*/(short)0, c, /*reuse_a=*/false, /*reuse_b=*/false);
        }
    }

    // Row-norm partial: lanes l and l^16 hold the two K-phase halves of row n16.
    nrm += __shfl_xor(nrm, 16, 32);
    if (lane < 16) {
        atomicAdd(&ws[NEG2_OFF + b * NROWS + g], nrm);
        // C rows 0..2 (the x rows) live in c[0..2] on lanes 0..15, column n16.
        const int base = ((b * NTILES + tile) * KDIM) * 16 + n16;
        atomicAdd(&ws[DOT_OFF + base + 0 * 16], c[0]);
        atomicAdd(&ws[DOT_OFF + base + 1 * 16], c[1]);
        atomicAdd(&ws[DOT_OFF + base + 2 * 16], c[2]);
    }
}

// ---------------- 3: masked logsumexp + final scalar ----------------
__global__ __launch_bounds__(32) void finalize_kernel(const int* __restrict__ gt_labels,
                                                      const int* __restrict__ gt_label_negs,
                                                      const float* __restrict__ ws,
                                                      float* __restrict__ out) {
    const int t = threadIdx.x;
    __shared__ float sl[32];
    float loss = 0.f;
    if (t < BDIM * KDIM) {
        const int b = t / KDIM;
        const int k = t % KDIM;
        const float na = sqrtf(ws[XN2_OFF + t]);
        const float npn = sqrtf(ws[XPN2_OFF + t]);
        const float pos = ws[XDOT_OFF + t] / fmaxf(na * npn, EPSV) * TEMP_INV;
        const int lbl = gt_labels[b * JDIM + (JDIM - 1)];

        float vals[NDIM * MDIM];
        float mx = pos;
        for (int nm = 0; nm < NDIM * MDIM; ++nm) {
            float v = -INFINITY;
            if (gt_label_negs[b * (NDIM * MDIM) + nm] == lbl) {
                const int g = nm * KDIM + k;
                const int tl = g >> 4, col = g & 15;
                const float dot = ws[DOT_OFF + ((b * NTILES + tl) * KDIM + k) * 16 + col];
                const float nn = sqrtf(ws[NEG2_OFF + b * NROWS + g]);
                v = dot / fmaxf(na * nn, EPSV) * TEMP_INV;
                mx = fmaxf(mx, v);
            }
            vals[nm] = v;
        }
        float s = expf(pos - mx);
        for (int nm = 0; nm < NDIM * MDIM; ++nm) s += expf(vals[nm] - mx);  // exp(-inf)=0
        loss = (mx + logf(s)) - pos;
    }
    sl[t] = loss;
    __syncthreads();
    if (t == 0) {
        float tot = 0.f;
        for (int i = 0; i < BDIM * KDIM; ++i) tot += sl[i];
        out[0] = tot / (2.0f * BDIM);
    }
}

// ---------------- launcher ----------------
extern "C" void kernel_launch(void* const* d_in, const int* in_sizes, int n_in,
                              void* d_out, int out_size, void* d_ws, size_t ws_size,
                              hipStream_t stream) {
    const float* fmaps     = (const float*)d_in[0];
    const float* fmaps_pos = (const float*)d_in[1];
    const float* fmaps_neg = (const float*)d_in[2];
    const int* gt_labels     = (const int*)d_in[3];
    const int* gt_label_negs = (const int*)d_in[4];
    float* ws  = (float*)d_ws;
    float* out = (float*)d_out;

    zero_ws_kernel<<<1, 256, 0, stream>>>(ws);
    pos_stats_kernel<<<BDIM * KDIM, 256, 0, stream>>>(fmaps, fmaps_pos, ws);
    neg_gemm_kernel<<<dim3(BDIM, NTILES, 16), 128, 0, stream>>>(fmaps, fmaps_neg, ws);
    finalize_kernel<<<1, 32, 0, stream>>>(gt_labels, gt_label_negs, ws, out);
}